// CoupledFEMSolver_43087111914309
// MI455X (gfx1250) — compile-verified
//
#include <hip/hip_runtime.h>

typedef __attribute__((ext_vector_type(2))) float v2f;
typedef __attribute__((ext_vector_type(8))) float v8f;

__device__ __forceinline__ void atomic_add_f32(float* p, float v) {
    __hip_atomic_fetch_add(p, v, __ATOMIC_RELAXED, __HIP_MEMORY_SCOPE_AGENT);
}

// ---------------------------------------------------------------------------
// Tet geometry: shape-function gradients + volume (analytic 3x3 cofactor inv)
// grads match reference A_inv construction exactly:
//   g1 = cross(v1,v2)/det, g2 = cross(v2,v0)/det, g3 = cross(v0,v1)/det,
//   g0 = -(g1+g2+g3),  vol = |det|/6
// ---------------------------------------------------------------------------
struct TetGrads {
    float g0x, g0y, g0z;
    float g1x, g1y, g1z;
    float g2x, g2y, g2z;
    float g3x, g3y, g3z;
    float vol;
};

__device__ __forceinline__ TetGrads tet_grads(const float* __restrict__ nodes,
                                              int n0, int n1, int n2, int n3) {
    const float p0x = nodes[n0 * 3 + 0], p0y = nodes[n0 * 3 + 1], p0z = nodes[n0 * 3 + 2];
    const float p1x = nodes[n1 * 3 + 0], p1y = nodes[n1 * 3 + 1], p1z = nodes[n1 * 3 + 2];
    const float p2x = nodes[n2 * 3 + 0], p2y = nodes[n2 * 3 + 1], p2z = nodes[n2 * 3 + 2];
    const float p3x = nodes[n3 * 3 + 0], p3y = nodes[n3 * 3 + 1], p3z = nodes[n3 * 3 + 2];

    const float v0x = p1x - p0x, v0y = p1y - p0y, v0z = p1z - p0z;
    const float v1x = p2x - p0x, v1y = p2y - p0y, v1z = p2z - p0z;
    const float v2x = p3x - p0x, v2y = p3y - p0y, v2z = p3z - p0z;

    const float c12x = v1y * v2z - v1z * v2y;
    const float c12y = v1z * v2x - v1x * v2z;
    const float c12z = v1x * v2y - v1y * v2x;
    const float det = v0x * c12x + v0y * c12y + v0z * c12z;
    const float inv = 1.0f / det;

    TetGrads t;
    t.g1x = c12x * inv; t.g1y = c12y * inv; t.g1z = c12z * inv;
    t.g2x = (v2y * v0z - v2z * v0y) * inv;
    t.g2y = (v2z * v0x - v2x * v0z) * inv;
    t.g2z = (v2x * v0y - v2y * v0x) * inv;
    t.g3x = (v0y * v1z - v0z * v1y) * inv;
    t.g3y = (v0z * v1x - v0x * v1z) * inv;
    t.g3z = (v0x * v1y - v0y * v1x) * inv;
    t.g0x = -(t.g1x + t.g2x + t.g3x);
    t.g0y = -(t.g1y + t.g2y + t.g3y);
    t.g0z = -(t.g1z + t.g2z + t.g3z);
    t.vol = fabsf(det) * (1.0f / 6.0f);
    return t;
}

// ---------------------------------------------------------------------------
// Zero-fill output (336 MB) with float4 stores, grid-stride.
// ---------------------------------------------------------------------------
__global__ void fem_zero_kernel(float* __restrict__ out, long n) {
    const long stride = (long)gridDim.x * blockDim.x * 4;
    long i = ((long)blockIdx.x * blockDim.x + threadIdx.x) * 4;
    const float4 z = make_float4(0.f, 0.f, 0.f, 0.f);
    for (; i + 3 < n; i += stride) {
        *reinterpret_cast<float4*>(out + i) = z;
    }
    if (blockIdx.x == 0 && threadIdx.x == 0) {
        for (long t = n & ~3L; t < n; ++t) out[t] = 0.f;
    }
}

// ---------------------------------------------------------------------------
// Fluid assembly: one wave handles 4 tets. K_e/vol = G*G^T computed with a
// single V_WMMA_F32_16X16X4_F32: rows = 16 gradient vectors (4 elems x 4
// nodes), K padded 3 -> 4. Diagonal 4x4 blocks of the 16x16 product are the
// per-element K_e; cross-element blocks are discarded.
//
// A layout (32-bit, 16x4): lanes 0-15 -> M=lane, vgpr0=K0, vgpr1=K1;
//                          lanes 16-31 -> M=lane-16, vgpr0=K2, vgpr1=K3.
// By symmetry the B fragment is identical per lane.
// C layout: vgpr r, lanes 0-15 -> (M=r,   N=lane),
//                   lanes 16-31 -> (M=r+8, N=lane-16).
// ---------------------------------------------------------------------------
__global__ void fluid_wmma_kernel(const float* __restrict__ nodes,
                                  const int* __restrict__ elems,
                                  float* __restrict__ Kf,
                                  float* __restrict__ Mf,
                                  int n_elems, int n_nodes) {
    const int lane = threadIdx.x & 31;
    const long gwave = ((long)blockIdx.x * blockDim.x + threadIdx.x) >> 5;

    const int m = lane & 15;   // global row / column id 0..15
    const int hi = lane >> 4;  // lane-group (K half for A/B, M+8 for C)
    const int e = m >> 2;      // element slot within the wave's 4 elements
    const int a = m & 3;       // node within element

    const long eid = gwave * 4 + e;
    const bool valid = eid < (long)n_elems;
    const long ec = valid ? eid : 0;

    const int n0 = elems[ec * 4 + 0];
    const int n1 = elems[ec * 4 + 1];
    const int n2 = elems[ec * 4 + 2];
    const int n3 = elems[ec * 4 + 3];

    const TetGrads t = tet_grads(nodes, n0, n1, n2, n3);

    // gradient row for this lane's (element, node)
    const float gax = (a == 0) ? t.g0x : (a == 1) ? t.g1x : (a == 2) ? t.g2x : t.g3x;
    const float gay = (a == 0) ? t.g0y : (a == 1) ? t.g1y : (a == 2) ? t.g2y : t.g3y;
    const float gaz = (a == 0) ? t.g0z : (a == 1) ? t.g1z : (a == 2) ? t.g2z : t.g3z;

    v2f ab;
    ab.x = hi ? gaz : gax;
    ab.y = hi ? 0.0f : gay;

    v8f c = {0.f, 0.f, 0.f, 0.f, 0.f, 0.f, 0.f, 0.f};
    // EXEC is all-1s here (no divergence above); 8-arg VOP3P form.
    c = __builtin_amdgcn_wmma_f32_16x16x4_f32(false, ab, false, ab,
                                              (short)0, c, false, false);

    if (valid) {
        // This lane owns column N = m of the 16x16 product; its element block
        // is e (same as the row-setup element). b = column node.
        const int b = a;  // N & 3 == m & 3
        const int coln = (b == 0) ? n0 : (b == 1) ? n1 : (b == 2) ? n2 : n3;
        const float vol = t.vol;
#pragma unroll
        for (int r = 0; r < 8; ++r) {
            const int j = r + 8 * hi - 4 * e;  // row-node within the block
            if (j >= 0 && j < 4) {
                const int rown = (j == 0) ? n0 : (j == 1) ? n1 : (j == 2) ? n2 : n3;
                const float kv = vol * c[r];
                const float mv = vol * ((j == b) ? 0.3f : 0.1f);
                const long off = (long)rown * n_nodes + coln;
                atomic_add_f32(Kf + off, kv);
                atomic_add_f32(Mf + off, mv);
            }
        }
    }
}

// ---------------------------------------------------------------------------
// Solid assembly: 16 threads per element, one per node pair (a,b).
// K block (3x3): vol * ( lam*ga_i*gb_j + mu*ga_j*gb_i + delta_ij*mu*(ga.gb) )
// M: lumped diagonal rho*vol/4 per dof (handled by a==b threads).
// ---------------------------------------------------------------------------
__global__ void solid_kernel(const float* __restrict__ nodes,
                             const int* __restrict__ elems,
                             const float* __restrict__ pE,
                             const float* __restrict__ pNu,
                             const float* __restrict__ pRho,
                             float* __restrict__ Ks,
                             float* __restrict__ Ms,
                             int n_elems, int n_nodes) {
    const long tid = (long)blockIdx.x * blockDim.x + threadIdx.x;
    const long eid = tid >> 4;
    if (eid >= (long)n_elems) return;
    const int pair = (int)(tid & 15);
    const int a = pair >> 2;
    const int b = pair & 3;

    const int n0 = elems[eid * 4 + 0];
    const int n1 = elems[eid * 4 + 1];
    const int n2 = elems[eid * 4 + 2];
    const int n3 = elems[eid * 4 + 3];

    const TetGrads t = tet_grads(nodes, n0, n1, n2, n3);

    const float E = pE[0];
    const float nu = pNu[0];
    const float rho = pRho[0];
    const float coeff = E / ((1.0f + nu) * (1.0f - 2.0f * nu));
    const float lam = coeff * nu;
    const float mu = coeff * (1.0f - 2.0f * nu) * 0.5f;

    const float gax = (a == 0) ? t.g0x : (a == 1) ? t.g1x : (a == 2) ? t.g2x : t.g3x;
    const float gay = (a == 0) ? t.g0y : (a == 1) ? t.g1y : (a == 2) ? t.g2y : t.g3y;
    const float gaz = (a == 0) ? t.g0z : (a == 1) ? t.g1z : (a == 2) ? t.g2z : t.g3z;
    const float gbx = (b == 0) ? t.g0x : (b == 1) ? t.g1x : (b == 2) ? t.g2x : t.g3x;
    const float gby = (b == 0) ? t.g0y : (b == 1) ? t.g1y : (b == 2) ? t.g2y : t.g3y;
    const float gbz = (b == 0) ? t.g0z : (b == 1) ? t.g1z : (b == 2) ? t.g2z : t.g3z;

    const int na = (a == 0) ? n0 : (a == 1) ? n1 : (a == 2) ? n2 : n3;
    const int nb = (b == 0) ? n0 : (b == 1) ? n1 : (b == 2) ? n2 : n3;

    const float ga[3] = {gax, gay, gaz};  // constant-indexed after unroll
    const float gb[3] = {gbx, gby, gbz};
    const float dotab = gax * gbx + gay * gby + gaz * gbz;
    const float vol = t.vol;

    const long W = 3L * n_nodes;
    const long rb = 3L * na;
    const long cb = 3L * nb;

#pragma unroll
    for (int i = 0; i < 3; ++i) {
#pragma unroll
        for (int j = 0; j < 3; ++j) {
            const float v = vol * (lam * ga[i] * gb[j] + mu * ga[j] * gb[i] +
                                   ((i == j) ? mu * dotab : 0.0f));
            atomic_add_f32(Ks + (rb + i) * W + (cb + j), v);
        }
    }
    if (a == b) {
        const float mv = rho * vol * 0.25f;
#pragma unroll
        for (int i = 0; i < 3; ++i) {
            atomic_add_f32(Ms + (rb + i) * W + (rb + i), mv);
        }
    }
}

// ---------------------------------------------------------------------------
// Host launch
// ---------------------------------------------------------------------------
extern "C" void kernel_launch(void* const* d_in, const int* in_sizes, int n_in,
                              void* d_out, int out_size, void* d_ws, size_t ws_size,
                              hipStream_t stream) {
    (void)n_in; (void)d_ws; (void)ws_size;

    const float* nodes = (const float*)d_in[0];
    const int* fe = (const int*)d_in[1];
    const int* se = (const int*)d_in[2];
    const float* pE = (const float*)d_in[3];
    const float* pNu = (const float*)d_in[4];
    const float* pRho = (const float*)d_in[5];

    const int n_nodes = in_sizes[0] / 3;
    const int n_fluid = in_sizes[1] / 4;
    const int n_solid = in_sizes[2] / 4;

    float* out = (float*)d_out;
    const size_t nn = (size_t)n_nodes * (size_t)n_nodes;
    const size_t n3 = 3 * (size_t)n_nodes;
    float* Kf = out;
    float* Mf = out + nn;
    float* Ks = out + 2 * nn;
    float* Ms = out + 2 * nn + n3 * n3;

    // 1) zero 336 MB of output
    fem_zero_kernel<<<4096, 256, 0, stream>>>(out, (long)out_size);

    // 2) fluid: 4 elements per wave, 8 waves per block (wave32)
    const int waves = (n_fluid + 3) / 4;
    const int fblocks = (waves + 7) / 8;
    fluid_wmma_kernel<<<fblocks, 256, 0, stream>>>(nodes, fe, Kf, Mf, n_fluid, n_nodes);

    // 3) solid: 16 threads per element
    const long sthreads = 16L * (long)n_solid;
    const int sblocks = (int)((sthreads + 255) / 256);
    solid_kernel<<<sblocks, 256, 0, stream>>>(nodes, se, pE, pNu, pRho, Ks, Ms,
                                              n_solid, n_nodes);
}